// MultiLayerGnn_41790031790095
// MI455X (gfx1250) — compile-verified
//
#include <hip/hip_runtime.h>

// ---------------------------------------------------------------------------
// MultiLayerGnn for MI455X (gfx1250, wave32).
// - GEMMs: V_WMMA_F32_16X16X4_F32, one wave per 16-row strip covering all P
//   columns. W is staged *transposed* in LDS so every B fragment is a single
//   ds_load_b64 with an immediate offset; A fragments are global_load_b64
//   with immediate offsets. fp32 kept end-to-end: the workload is
//   scatter/atomic bound (~1 GB traffic ~= 45us @ 23.3 TB/s vs 1.8 GFLOP of
//   GEMM), so precision is free.
// - Top-K: all 50000 sort keys staged once per block in LDS (200 KB fits the
//   320 KB/WGP LDS); exact stable rank reproduces argsort(-h[:,-1]).
// ---------------------------------------------------------------------------

#define N_NODES 50000
#define N_EDGES 800000
#define COUT    32
#define TOPK    1024

typedef __attribute__((ext_vector_type(2))) float v2f;
typedef __attribute__((ext_vector_type(8))) float v8f;

__global__ void zero_kernel(float* __restrict__ p, int n) {
  int i = blockIdx.x * blockDim.x + threadIdx.x;
  if (i < n) p[i] = 0.0f;
}

__global__ void deg_kernel(const int* __restrict__ dst, float* __restrict__ deg) {
  int e = blockIdx.x * blockDim.x + threadIdx.x;
  if (e < N_EDGES) atomicAdd(&deg[dst[e]], 1.0f);
}

__global__ void dinv_kernel(float* __restrict__ deg) {
  int i = blockIdx.x * blockDim.x + threadIdx.x;
  if (i < N_NODES) deg[i] = rsqrtf(deg[i] + 1.0f);   // +1 = self loop; deg>=1
}

// C[N,P] = A[N,64] @ W[64,P].  One wave per 16-row strip (P/16 WMMA tiles).
// A-frag (ISA 7.12.2, 32-bit A 16x4): lane<16 -> K=k0,k0+1 of row M=lane;
// lane>=16 -> K=k0+2,k0+3 of row M=lane-16.  B staged transposed in LDS so
// b = (W[k][n], W[k+1][n]) is one contiguous ds_load_b64.
template <int P>
__global__ void gemm_wmma(const float* __restrict__ A, const float* __restrict__ W,
                          float* __restrict__ C, int n_strips) {
  __shared__ float wt[P * 64];                 // wt[n][k] = W[k][n]
  for (int idx = threadIdx.x; idx < 64 * P; idx += blockDim.x) {
    int k = idx / P, n = idx - k * P;
    wt[n * 64 + k] = W[idx];
  }
  __syncthreads();

  int wave = (int)((blockIdx.x * blockDim.x + threadIdx.x) >> 5);
  if (wave >= n_strips) return;                // wave-uniform: EXEC all-1s
  int lane = threadIdx.x & 31;
  int m    = lane & 15;
  int koff = (lane >> 4) << 1;                 // 0 or 2
  const float* arow = A + (size_t)(wave * 16 + m) * 64;
  const float* wrow = wt + m * 64 + koff;      // + t*16*64 + k0 are immediates

  v8f acc[P / 16];
#pragma unroll
  for (int t = 0; t < P / 16; ++t) acc[t] = (v8f){};

#pragma unroll
  for (int k0 = 0; k0 < 64; k0 += 4) {
    v2f a = *(const v2f*)(arow + k0 + koff);   // 8B-aligned immediate offset
#pragma unroll
    for (int t = 0; t < P / 16; ++t) {
      v2f b = *(const v2f*)(wrow + t * 16 * 64 + k0);   // ds_load_b64 imm
      acc[t] = __builtin_amdgcn_wmma_f32_16x16x4_f32(
          false, a, false, b, (short)0, acc[t], false, false);
    }
  }

  int half = (lane >> 4) << 3;                 // 0 or 8
  float* crow = C + (size_t)(wave * 16 + half) * P + m;
#pragma unroll
  for (int t = 0; t < P / 16; ++t)
#pragma unroll
    for (int r = 0; r < 8; ++r)
      crow[(size_t)r * P + 16 * t] = acc[t][r];
}

// agg[dst] += dinv[src]*dinv[dst] * HW[src]   (4 channels per thread)
__global__ void scatter_kernel(const float* __restrict__ HW, const int* __restrict__ src,
                               const int* __restrict__ dst, const float* __restrict__ dinv,
                               float* __restrict__ agg, int P) {
  int tid = blockIdx.x * blockDim.x + threadIdx.x;
  int per_edge = P >> 2;
  int e = tid / per_edge;
  if (e >= N_EDGES) return;
  int c = (tid - e * per_edge) << 2;
  int s = src[e], d = dst[e];
  float w = dinv[s] * dinv[d];
  const float* hp = HW + (size_t)s * P + c;
  float* ap = agg + (size_t)d * P + c;
  atomicAdd(ap + 0, w * hp[0]);
  atomicAdd(ap + 1, w * hp[1]);
  atomicAdd(ap + 2, w * hp[2]);
  atomicAdd(ap + 3, w * hp[3]);
}

// h = relu(agg + dinv^2*HW + b + 0.5*(XL + bl))       (P = 64)
__global__ void combine_relu_kernel(const float* __restrict__ agg, const float* __restrict__ HW,
                                    const float* __restrict__ XL, const float* __restrict__ dinv,
                                    const float* __restrict__ b, const float* __restrict__ bl,
                                    float* __restrict__ h) {
  int tid = blockIdx.x * blockDim.x + threadIdx.x;
  if (tid >= N_NODES * 64) return;
  int i = tid >> 6, c = tid & 63;
  float di = dinv[i];
  float v = agg[tid] + di * di * HW[tid] + b[c] + 0.5f * (XL[tid] + bl[c]);
  h[tid] = v > 0.0f ? v : 0.0f;
}

// h3 = agg + dinv^2*HW + b   (P = 32); channel 31 also emits an order key.
__global__ void combine3_kernel(const float* __restrict__ agg, const float* __restrict__ HW,
                                const float* __restrict__ dinv, const float* __restrict__ b,
                                float* __restrict__ h3, unsigned* __restrict__ keys) {
  int tid = blockIdx.x * blockDim.x + threadIdx.x;
  if (tid >= N_NODES * 32) return;
  int i = tid >> 5, c = tid & 31;
  float di = dinv[i];
  float v = agg[tid] + di * di * HW[tid] + b[c];
  h3[tid] = v;
  if (c == 31) {
    unsigned u = __float_as_uint(v);           // monotone float->uint map
    u = (u & 0x80000000u) ? ~u : (u | 0x80000000u);
    keys[i] = u;
  }
}

// Stable top-K: each block stages ALL keys in LDS (200 KB < 320 KB/WGP),
// each thread computes its node's exact rank (desc value, asc index).
__global__ void topk_kernel(const unsigned* __restrict__ keys,
                            const float* __restrict__ h3, float* __restrict__ out) {
  extern __shared__ unsigned s_keys[];
  for (int j = threadIdx.x; j < N_NODES; j += blockDim.x) s_keys[j] = keys[j];
  __syncthreads();
  int i = blockIdx.x * blockDim.x + threadIdx.x;
  if (i >= N_NODES) return;
  unsigned my = s_keys[i];
  int rank = 0;
  for (int j = 0; j < N_NODES; ++j) {          // j wave-uniform: LDS broadcast
    unsigned kj = s_keys[j];
    rank += (int)((kj > my) | ((kj == my) & (j < i)));
  }
  if (rank < TOPK) {
    const float* sp = h3 + (size_t)i * COUT;
    float* dp = out + (size_t)rank * COUT;
#pragma unroll
    for (int c = 0; c < COUT; ++c) dp[c] = sp[c];
  }
}

extern "C" void kernel_launch(void* const* d_in, const int* in_sizes, int n_in,
                              void* d_out, int out_size, void* d_ws, size_t ws_size,
                              hipStream_t stream) {
  (void)in_sizes; (void)n_in; (void)out_size; (void)ws_size;
  const float* x   = (const float*)d_in[0];
  const float* W1  = (const float*)d_in[1];
  const float* b1  = (const float*)d_in[2];
  const float* Wl1 = (const float*)d_in[3];
  const float* bl1 = (const float*)d_in[4];
  const float* W2  = (const float*)d_in[5];
  const float* b2  = (const float*)d_in[6];
  const float* Wl2 = (const float*)d_in[7];
  const float* bl2 = (const float*)d_in[8];
  const float* W3  = (const float*)d_in[9];
  const float* b3  = (const float*)d_in[10];
  const int*   ei  = (const int*)d_in[11];     // edge_index [2,E]
  const int* esrc = ei;
  const int* edst = ei + N_EDGES;
  float* out = (float*)d_out;

  float* ws   = (float*)d_ws;
  float* buf0 = ws;                                  // HW   [N,64]
  float* buf1 = ws + (size_t)N_NODES * 64;           // XL / h3
  float* buf2 = ws + (size_t)N_NODES * 128;          // agg
  float* buf3 = ws + (size_t)N_NODES * 192;          // h (layer in/out)
  float* dinv = ws + (size_t)N_NODES * 256;          // deg -> dinv
  unsigned* keys = (unsigned*)(dinv + N_NODES);

  const int B = 256;
  auto g = [](long long n, int b) { return (int)((n + b - 1) / b); };

  // degree / normalization
  zero_kernel<<<g(N_NODES, B), B, 0, stream>>>(dinv, N_NODES);
  deg_kernel<<<g(N_EDGES, B), B, 0, stream>>>(edst, dinv);
  dinv_kernel<<<g(N_NODES, B), B, 0, stream>>>(dinv);

  const int n_strips = N_NODES / 16;                 // 3125, exact
  const int gemm_blocks = g((long long)n_strips * 32, B);

  // ---- layer 1 ----
  gemm_wmma<64><<<gemm_blocks, B, 0, stream>>>(x, W1,  buf0, n_strips);
  gemm_wmma<64><<<gemm_blocks, B, 0, stream>>>(x, Wl1, buf1, n_strips);
  zero_kernel<<<g((long long)N_NODES * 64, B), B, 0, stream>>>(buf2, N_NODES * 64);
  scatter_kernel<<<g((long long)N_EDGES * 16, B), B, 0, stream>>>(buf0, esrc, edst, dinv, buf2, 64);
  combine_relu_kernel<<<g((long long)N_NODES * 64, B), B, 0, stream>>>(buf2, buf0, buf1, dinv, b1, bl1, buf3);

  // ---- layer 2 ----
  gemm_wmma<64><<<gemm_blocks, B, 0, stream>>>(buf3, W2,  buf0, n_strips);
  gemm_wmma<64><<<gemm_blocks, B, 0, stream>>>(buf3, Wl2, buf1, n_strips);
  zero_kernel<<<g((long long)N_NODES * 64, B), B, 0, stream>>>(buf2, N_NODES * 64);
  scatter_kernel<<<g((long long)N_EDGES * 16, B), B, 0, stream>>>(buf0, esrc, edst, dinv, buf2, 64);
  combine_relu_kernel<<<g((long long)N_NODES * 64, B), B, 0, stream>>>(buf2, buf0, buf1, dinv, b2, bl2, buf3);

  // ---- layer 3 ----
  gemm_wmma<32><<<gemm_blocks, B, 0, stream>>>(buf3, W3, buf0, n_strips);
  zero_kernel<<<g((long long)N_NODES * 32, B), B, 0, stream>>>(buf2, N_NODES * 32);
  scatter_kernel<<<g((long long)N_EDGES * 8, B), B, 0, stream>>>(buf0, esrc, edst, dinv, buf2, 32);
  combine3_kernel<<<g((long long)N_NODES * 32, B), B, 0, stream>>>(buf2, buf0, dinv, b3, buf1, keys);

  // ---- stable top-K (global_sort_pool) ----
  const size_t smem = (size_t)N_NODES * sizeof(unsigned);  // 200 KB LDS
  topk_kernel<<<g(N_NODES, 1024), 1024, smem, stream>>>(keys, buf1, out);
}